// Criterion_73203422593120
// MI455X (gfx1250) — compile-verified
//
#include <hip/hip_runtime.h>
#include <math.h>

// Problem dims (from reference)
#define L_  3
#define K_  6
#define T_  80
#define B_  256
#define KG_ 64
#define NO_ 192
#define T2_ 80

#define ENTROPY_W 40.0f
#define KL_W      20.0f
#define GOAL_REG_W 1.0f
#define DISP_W     0.5f
#define LOG_STD_LO (-1.609f)
#define LOG_STD_HI (5.0f)
#define RHO_LIM    (0.5f)
#define LOG_2PI_F  (1.8378770664093453f)

typedef __attribute__((ext_vector_type(2))) float v2f;
typedef __attribute__((ext_vector_type(8))) float v8f;

__device__ __forceinline__ float waveSum(float v) {
#pragma unroll
  for (int off = 16; off > 0; off >>= 1) v += __shfl_xor(v, off, 32);
  return v;
}
__device__ __forceinline__ float waveMin(float v) {
#pragma unroll
  for (int off = 16; off > 0; off >>= 1) v = fminf(v, __shfl_xor(v, off, 32));
  return v;
}

// ---------------------------------------------------------------------------
// Layer loss: one wave per (l,b). Streams preds[l,:,:,b,:] once.
// ws_layer[l*B+b] = per-(l,b) contribution already scaled by 1/(B*L).
// ---------------------------------------------------------------------------
__global__ __launch_bounds__(32) void layer_kernel(
    const float* __restrict__ modes,   // (L,B,K)
    const float* __restrict__ preds,   // (L,K,T,B,5)
    const float* __restrict__ gt,      // (B,T,5)
    float* __restrict__ ws_layer) {
  const int lb = blockIdx.x;
  const int l = lb / B_;
  const int b = lb % B_;
  const int lane = threadIdx.x;

  float nllk[K_], entk[K_], adk[K_];

#pragma unroll
  for (int k = 0; k < K_; ++k) {
    float pn = 0.f, pe = 0.f, pa = 0.f;
    const float* pk = preds + ((size_t)(l * K_ + k)) * T_ * B_ * 5 + (size_t)b * 5;
    for (int t = lane; t < T_; t += 32) {
      const float* p = pk + (size_t)t * B_ * 5;
      const float* g = gt + ((size_t)b * T_ + t) * 5;
      float mux = p[0], muy = p[1], bx = p[2], by = p[3], rho = p[4];
      float gx = g[0], gy = g[1], m = g[4];
      float ax = fabsf(gx - mux), ay = fabsf(gy - muy);
      // Laplace NLL, masked, summed over t
      pn += m * (logf(2.f * bx) + ax / bx + logf(2.f * by) + ay / by);
      // entropy term (no mask), summed over t
      float sxsy = bx * by;
      pe += 1.f + LOG_2PI_F + 0.5f * logf(sxsy * sxsy * (1.f - rho * rho));
      // masked L2 for ADE
      pa += m * sqrtf(ax * ax + ay * ay);
    }
    pn = waveSum(pn); pe = waveSum(pe); pa = waveSum(pa);
    // FDE at t = T-1 (wave-uniform scalar compute)
    const float* pl = pk + (size_t)(T_ - 1) * B_ * 5;
    const float* gl = gt + ((size_t)b * T_ + (T_ - 1)) * 5;
    float dx = pl[0] - gl[0], dy = pl[1] - gl[1];
    float fde = sqrtf(dx * dx + dy * dy) * gl[4];
    nllk[k] = pn;
    entk[k] = pe;
    adk[k]  = fde + pa / (float)T_;
  }

  // posterior softmax over K, KL, entropy max, min(ade+fde)
  float z[K_], e[K_];
  float zmax = -__builtin_inff();
#pragma unroll
  for (int k = 0; k < K_; ++k) {
    float lm = logf(modes[(size_t)l * B_ * K_ + (size_t)b * K_ + k]);
    z[k] = -nllk[k] + lm;
    zmax = fmaxf(zmax, z[k]);
  }
  float s = 0.f;
#pragma unroll
  for (int k = 0; k < K_; ++k) { e[k] = expf(z[k] - zmax); s += e[k]; }
  float logs = logf(s);
  float loss = 0.f, kl = 0.f;
  float entmax = -__builtin_inff(), admin = __builtin_inff();
#pragma unroll
  for (int k = 0; k < K_; ++k) {
    float post = e[k] / s;
    float logpost = z[k] - zmax - logs;
    float logm = logf(modes[(size_t)l * B_ * K_ + (size_t)b * K_ + k]);
    loss += nllk[k] * post;
    kl += post * (logpost - logm);
    entmax = fmaxf(entmax, entk[k]);
    admin = fminf(admin, adk[k]);
  }
  float contrib = (loss + ENTROPY_W * entmax + KL_W * kl + 100.f * admin)
                  / (float)(B_ * L_);
  if (lane == 0) ws_layer[lb] = contrib;
}

// ---------------------------------------------------------------------------
// Goal loss: one wave per (l,b); lanes cover KG=64 goals (2 each).
// ---------------------------------------------------------------------------
__global__ __launch_bounds__(32) void goal_kernel(
    const float* __restrict__ goal_reg,  // (L,KG,B,2)
    const float* __restrict__ goal_fde,  // (L,B,KG)
    const float* __restrict__ gt,        // (B,T,5)
    float* __restrict__ ws_goal) {
  const int lb = blockIdx.x;
  const int l = lb / B_;
  const int b = lb % B_;
  const int lane = threadIdx.x;

  const float* gl = gt + ((size_t)b * T_ + (T_ - 1)) * 5;
  const float gx = gl[0], gy = gl[1], m = gl[4];

  float minreg = __builtin_inff();
  float hub = 0.f;
  for (int kg = lane; kg < KG_; kg += 32) {
    const float* gr = goal_reg + (((size_t)l * KG_ + kg) * B_ + b) * 2;
    float dx = gr[0] - gx, dy = gr[1] - gy;
    float reg = sqrtf(dx * dx + dy * dy) * m;   // == fde_gt (stop_grad only)
    minreg = fminf(minreg, reg);
    float fd = goal_fde[(size_t)l * B_ * KG_ + (size_t)b * KG_ + kg];
    float d = fabsf(fd - reg);
    hub += (d < 1.f) ? 0.5f * d * d : d - 0.5f;   // smooth-L1
  }
  minreg = waveMin(minreg);
  hub = waveSum(hub);
  if (lane == 0)
    ws_goal[lb] = GOAL_REG_W * minreg / (float)B_
                + DISP_W * hub / (float)(B_ * KG_);
}

// ---------------------------------------------------------------------------
// Dense GMM loss: one block (8 waves) per b. Waves stride over NO=192 agents,
// lanes over T2=80 (28B contiguous per lane -> coalesced streaming of 188MB).
// ---------------------------------------------------------------------------
__global__ __launch_bounds__(256) void dense_kernel(
    const float* __restrict__ pred,    // (B,NO,T2,7)
    const float* __restrict__ state,   // (B,NO,T2,4)
    const float* __restrict__ mask,    // (B,NO,T2)
    float* __restrict__ ws_dense) {
  const int b = blockIdx.x;
  const int tid = threadIdx.x;
  const int wave = tid >> 5, lane = tid & 31;
  __shared__ float s_rv[8], s_v[8];

  float acc_rv = 0.f, acc_v = 0.f;
  for (int no = wave; no < NO_; no += 8) {
    const size_t base = (size_t)b * NO_ + no;
    const float* pp = pred + base * T2_ * 7;
    const float* ss = state + base * T2_ * 4;
    const float* mm = mask + base * T2_;
    float preg = 0.f, pm = 0.f;
    for (int t = lane; t < T2_; t += 32) {
      const float* p = pp + (size_t)t * 7;
      const float* st = ss + (size_t)t * 4;
      __builtin_prefetch(p + 32 * 7, 0, 1);    // global_prefetch_b8, next chunk
      __builtin_prefetch(st + 32 * 4, 0, 1);
      float m = mm[t];
      float p0 = p[0], p1 = p[1], p2 = p[2], p3 = p[3], p4 = p[4], p5 = p[5], p6 = p[6];
      float s0 = st[0], s1v = st[1], s2v = st[2], s3v = st[3];
      float vel = (fabsf(p5 - s2v) + fabsf(p6 - s3v)) * m;
      float dx = s0 - p0, dy = s1v - p1;
      float ls1 = fminf(fmaxf(p2, LOG_STD_LO), LOG_STD_HI);
      float ls2 = fminf(fmaxf(p3, LOG_STD_LO), LOG_STD_HI);
      float e1 = expf(ls1), e2 = expf(ls2);
      float rho = fminf(fmaxf(p4, -RHO_LIM), RHO_LIM);
      float om = 1.f - rho * rho;
      float coef = ls1 + ls2 + 0.5f * logf(om);
      float u = dx / e1, v = dy / e2;
      float expt = 0.5f / om * (u * u + v * v - 2.f * rho * u * v);
      preg += vel + (coef + expt) * m;
      pm += m;
    }
    preg = waveSum(preg);
    pm = waveSum(pm);
    float valid = (pm > 0.f) ? 1.f : 0.f;
    acc_rv += preg * valid;
    acc_v += valid;
  }
  if (lane == 0) { s_rv[wave] = acc_rv; s_v[wave] = acc_v; }
  __syncthreads();
  if (tid == 0) {
    float rv = 0.f, vv = 0.f;
    for (int w = 0; w < 8; ++w) { rv += s_rv[w]; vv += s_v[w]; }
    ws_dense[b] = (rv / fmaxf(vv, 1.f)) / (float)B_;   // mean over B folded in
  }
}

// ---------------------------------------------------------------------------
// Final reduction of the 1792 (= 28*64) partials on the matrix core:
// D = A x ones(4x16) + C sums 64 fp32 values per V_WMMA_F32_16X16X4_F32 into
// the fp32 accumulator in a fixed (deterministic) order.
// One full wave32, no divergence around WMMA (EXEC must be all ones).
// ---------------------------------------------------------------------------
__global__ __launch_bounds__(32) void reduce_kernel(
    const float* __restrict__ ws, float* __restrict__ out) {
  const int lane = threadIdx.x;
  v2f ones; ones.x = 1.f; ones.y = 1.f;
  v8f acc = {};
  const int NTOT = L_ * B_ + L_ * B_ + B_;  // 768 + 768 + 256 = 1792
  for (int base = 0; base < NTOT; base += 64) {
    v2f a;
    a.x = ws[base + 2 * lane + 0];
    a.y = ws[base + 2 * lane + 1];
    // D[m][n] = sum_k A[m][k] (B all-ones) + C  -> accumulates row sums
    acc = __builtin_amdgcn_wmma_f32_16x16x4_f32(
        false, a, false, ones, (short)0, acc, false, false);
  }
  // lanes 0-15 hold rows 0..7 (replicated over n), lanes 16-31 rows 8..15
  float s = 0.f;
#pragma unroll
  for (int j = 0; j < 8; ++j) s += acc[j];
  float hi = __shfl(s, 16, 32);
  if (lane == 0) out[0] = s + hi;
}

// ---------------------------------------------------------------------------
extern "C" void kernel_launch(void* const* d_in, const int* in_sizes, int n_in,
                              void* d_out, int out_size, void* d_ws, size_t ws_size,
                              hipStream_t stream) {
  (void)in_sizes; (void)n_in; (void)out_size; (void)ws_size;
  const float* modes    = (const float*)d_in[0];  // (L,B,K)
  const float* preds    = (const float*)d_in[1];  // (L,K,T,B,5)
  const float* gt       = (const float*)d_in[2];  // (B,T,5)
  const float* goal_reg = (const float*)d_in[3];  // (L,KG,B,2)
  const float* goal_fde = (const float*)d_in[4];  // (L,B,KG)
  const float* dpred    = (const float*)d_in[5];  // (B,NO,T2,7)
  const float* dstate   = (const float*)d_in[6];  // (B,NO,T2,4)
  const float* dmask    = (const float*)d_in[7];  // (B,NO,T2)

  float* ws = (float*)d_ws;
  float* ws_layer = ws;                    // 768 floats
  float* ws_goal  = ws + (size_t)L_ * B_;  // 768 floats
  float* ws_dense = ws + (size_t)2 * L_ * B_;  // 256 floats  (total 7 KB)

  layer_kernel<<<L_ * B_, 32, 0, stream>>>(modes, preds, gt, ws_layer);
  goal_kernel<<<L_ * B_, 32, 0, stream>>>(goal_reg, goal_fde, gt, ws_goal);
  dense_kernel<<<B_, 256, 0, stream>>>(dpred, dstate, dmask, ws_dense);
  reduce_kernel<<<1, 32, 0, stream>>>(ws, (float*)d_out);
}